// GCNModule_17978733101784
// MI455X (gfx1250) — compile-verified
//
#include <hip/hip_runtime.h>

#define N_NODES 6272
#define C_IN    2048
#define C_OUT   512
#define KNN     8
#define KC      32          // K-chunk staged in LDS (double-buffered)
#define GKC     32          // K-chunk for the layer GEMMs

typedef __attribute__((ext_vector_type(2))) float v2f;
typedef __attribute__((ext_vector_type(8))) float v8f;
typedef __attribute__((ext_vector_type(4))) unsigned int u32x4;
typedef __attribute__((ext_vector_type(4))) int i32x4;
typedef __attribute__((ext_vector_type(8))) int i32x8;

static __device__ __forceinline__ v8f vzero8() {
  v8f z = {0.f, 0.f, 0.f, 0.f, 0.f, 0.f, 0.f, 0.f};
  return z;
}

static __device__ __forceinline__ v8f wmma_f32(v2f a, v2f b, v8f c) {
  // D = A(16x4) * B(4x16) + C, fp32 everywhere (V_WMMA_F32_16X16X4_F32)
  return __builtin_amdgcn_wmma_f32_16x16x4_f32(
      /*neg_a=*/false, a, /*neg_b=*/false, b,
      /*c_mod=*/(short)0, c, /*reuse_a=*/false, /*reuse_b=*/false);
}

// ---- Tensor Data Mover: 2D f32 tile (tile_d0 x tile_d1) global->LDS.
// LDS rows are padded by 4 dwords every 32 dwords (pad_interval=4, pad_amount=3)
// to match the [row][KC+4] bank-conflict-free layout.
static __device__ __forceinline__ void tdm_load_2d(const void* gptr,
                                                   unsigned lds_addr,
                                                   unsigned tile_d0,
                                                   unsigned tile_d1,
                                                   unsigned tensor_d0,
                                                   unsigned tensor_d1,
                                                   unsigned stride0) {
  const unsigned long long ga = (unsigned long long)gptr;
  u32x4 g0;
  g0[0] = 1u;                                             // count=1 (valid)
  g0[1] = lds_addr;                                       // LDS byte address
  g0[2] = (unsigned)(ga & 0xffffffffu);                   // global_addr[31:0]
  g0[3] = (unsigned)((ga >> 32) & 0x01ffffffu)            // global_addr[56:32]
          | (2u << 30);                                   // type=2 (image/D#)
  i32x8 g1;
  g1[0] = (int)((2u << 16)                                // data_size = 4B
                | (1u << 20)                              // pad_enable
                | (4u << 22)                              // pad_interval: 32 dw
                | (3u << 25));                            // pad_amount : 4 dw
  g1[1] = (int)((tensor_d0 & 0xffffu) << 16);             // tensor_dim0 lo16
  g1[2] = (int)(((tensor_d0 >> 16) & 0xffffu)             // tensor_dim0 hi16
                | ((tensor_d1 & 0xffffu) << 16));         // tensor_dim1 lo16
  g1[3] = (int)(((tensor_d1 >> 16) & 0xffffu)             // tensor_dim1 hi16
                | ((tile_d0 & 0xffffu) << 16));           // tile_dim0
  g1[4] = (int)(tile_d1 & 0xffffu);                       // tile_dim1 (dim2=0)
  g1[5] = (int)stride0;                                   // dim0_stride[31:0]
  g1[6] = 0;                                              // stride0 hi / stride1 lo
  g1[7] = 0;
  i32x4 gz4 = {0, 0, 0, 0};                               // 2D: groups 2/3 unused
  i32x8 gz8 = {0, 0, 0, 0, 0, 0, 0, 0};
  __builtin_amdgcn_tensor_load_to_lds(g0, g1, gz4, gz4, gz8, 0);
}

static __device__ __forceinline__ unsigned lds_off(const void* p) {
  // LDS aperture: generic addr = {SHARED_BASE, offset[31:0]} -> low 32 bits
  return (unsigned)(unsigned long long)p;
}

// Insert (d, j) into ascending top-K kept entirely in VGPRs.
static __device__ __forceinline__ void topk_insert(float (&bd)[KNN], int (&bi)[KNN],
                                                   float d, int j) {
  if (d >= bd[KNN - 1]) return;
  int pos = 0;
#pragma unroll
  for (int s = 0; s < KNN; ++s) pos += (bd[s] <= d) ? 1 : 0;
#pragma unroll
  for (int p = KNN - 1; p > 0; --p)
    if (p > pos) { bd[p] = bd[p - 1]; bi[p] = bi[p - 1]; }
#pragma unroll
  for (int p = 0; p < KNN; ++p)
    if (p == pos) { bd[p] = d; bi[p] = j; }
}

// ---------------------------------------------------------------- row norms
__global__ void __launch_bounds__(256)
rowsq_kernel(const float* __restrict__ X, float* __restrict__ sq,
             float* __restrict__ deg) {
  __shared__ float red[256];
  const int i = blockIdx.x;
  const float* row = X + (size_t)i * C_IN;
  float s = 0.f;
  for (int c = threadIdx.x; c < C_IN; c += 256) { float v = row[c]; s += v * v; }
  red[threadIdx.x] = s;
  __syncthreads();
  for (int off = 128; off > 0; off >>= 1) {
    if (threadIdx.x < off) red[threadIdx.x] += red[threadIdx.x + off];
    __syncthreads();
  }
  if (threadIdx.x == 0) { sq[i] = red[0]; deg[i] = 1.0f; /* self loop */ }
}

// ---------------------------------------------------- fused gram + top-k KNN
// Block = 16 queries. Sweeps 49 panels of 128 columns (wave w owns 16 cols).
// A/B chunks DMA'd into double-buffered LDS by the TDM; one barrier per chunk;
// DMA of chunk g+1 overlaps the WMMAs of chunk g.
__global__ void __launch_bounds__(256)
knn_kernel(const float* __restrict__ X, const float* __restrict__ sq,
           int* __restrict__ knn_out, float* __restrict__ deg) {
  __shared__ __align__(16) float As[2][16][KC + 4];   //  4.6 KB
  __shared__ __align__(16) float Bs[2][128][KC + 4];  // 36.9 KB
  __shared__ float d2t[8][16][17];                    //  8.7 KB (padded)
  __shared__ float candD[8][16][KNN];                 //  4 KB
  __shared__ int   candI[8][16][KNN];                 //  4 KB

  const int tid   = threadIdx.x;
  const int wave  = tid >> 5;
  const int lane  = tid & 31;
  const int r     = lane & 15;
  const int hh    = lane >> 4;
  const int qbase = blockIdx.x * 16;

  const int CPT    = C_IN / KC;          // chunks per panel (64)
  const int NPAN   = N_NODES / 128;      // 49
  const int NCHUNK = NPAN * CPT;

  float bd[KNN]; int bi[KNN];
#pragma unroll
  for (int s = 0; s < KNN; ++s) { bd[s] = 3.0e38f; bi[s] = 0; }

  float sqm[8];
#pragma unroll
  for (int v = 0; v < 8; ++v) sqm[v] = sq[qbase + 8 * hh + v];

  // prologue: DMA chunk 0 into buffer 0
  if (wave == 0) {
    tdm_load_2d(X + (size_t)qbase * C_IN, lds_off(&As[0][0][0]),
                KC, 16, C_IN, N_NODES, C_IN);
    tdm_load_2d(X, lds_off(&Bs[0][0][0]),
                KC, 128, C_IN, N_NODES, C_IN);
    __builtin_amdgcn_s_wait_tensorcnt(0);
  }
  __syncthreads();

  for (int t = 0; t < NPAN; ++t) {
    const int ct = t * 128 + wave * 16;  // this wave's 16 columns
    v8f acc0 = vzero8(), acc1 = vzero8();

    for (int i = 0; i < CPT; ++i) {
      const int g   = t * CPT + i;
      const int buf = g & 1;
      if (wave == 0 && g + 1 < NCHUNK) { // DMA next chunk into idle buffer
        const int g2  = g + 1;
        const int t2  = g2 / CPT;
        const int kc2 = (g2 % CPT) * KC;
        tdm_load_2d(X + (size_t)qbase * C_IN + kc2, lds_off(&As[buf ^ 1][0][0]),
                    KC, 16, C_IN, N_NODES, C_IN);
        tdm_load_2d(X + (size_t)(t2 * 128) * C_IN + kc2, lds_off(&Bs[buf ^ 1][0][0]),
                    KC, 128, C_IN, N_NODES, C_IN);
      }
#pragma unroll
      for (int k4 = 0; k4 < KC; k4 += 8) {   // 8 WMMAs per staged chunk
        v2f a0 = *(const v2f*)&As[buf][r][k4 + 2 * hh];
        v2f b0 = *(const v2f*)&Bs[buf][wave * 16 + r][k4 + 2 * hh];
        acc0 = wmma_f32(a0, b0, acc0);
        v2f a1 = *(const v2f*)&As[buf][r][k4 + 4 + 2 * hh];
        v2f b1 = *(const v2f*)&Bs[buf][wave * 16 + r][k4 + 4 + 2 * hh];
        acc1 = wmma_f32(a1, b1, acc1);
      }
      if (wave == 0) __builtin_amdgcn_s_wait_tensorcnt(0);
      __syncthreads();
    }

    v8f dot = acc0 + acc1;
    const float sqn = sq[ct + r];
#pragma unroll
    for (int v = 0; v < 8; ++v) {
      const int m = 8 * hh + v;              // C layout: row = 8*half + vgpr
      d2t[wave][m][r] = sqm[v] + sqn - 2.0f * dot[v];
    }
    __syncthreads();
    if (lane < 16) {
#pragma unroll
      for (int n = 0; n < 16; ++n) {
        const int j = ct + n;
        const float d = d2t[wave][lane][n];
        if (j != qbase + lane) topk_insert(bd, bi, d, j);  // exclude self
      }
    }
    __syncthreads();
  }

  if (lane < 16) {
#pragma unroll
    for (int s = 0; s < KNN; ++s) {
      candD[wave][lane][s] = bd[s];
      candI[wave][lane][s] = bi[s];
    }
  }
  __syncthreads();

  if (wave == 0 && lane < 16) {              // merge 8 partial lists per row
    float fd[KNN]; int fi[KNN];
#pragma unroll
    for (int s = 0; s < KNN; ++s) { fd[s] = candD[0][lane][s]; fi[s] = candI[0][lane][s]; }
    for (int w = 1; w < 8; ++w) {
#pragma unroll
      for (int s = 0; s < KNN; ++s) topk_insert(fd, fi, candD[w][lane][s], candI[w][lane][s]);
    }
#pragma unroll
    for (int s = 0; s < KNN; ++s) {
      knn_out[(qbase + lane) * KNN + s] = fi[s];
      atomicAdd(deg + fi[s], 1.0f);          // in-degree from knn edges
    }
  }
}

// ----------------------------------------------------------------- deg->dinv
__global__ void __launch_bounds__(256) dinv_kernel(float* __restrict__ deg) {
  const int i = blockIdx.x * 256 + threadIdx.x;
  if (i < N_NODES) deg[i] = rsqrtf(fmaxf(deg[i], 1.0f));
}

// ------------------------------------------------- C[MxN] = A[MxK] * B[KxN]
// Block tile 128x32. A chunks DMA'd by TDM (double-buffered); transposed W
// chunk prefetched into registers during compute, stored to the idle buffer.
__global__ void __launch_bounds__(256)
gemm_wmma_kernel(const float* __restrict__ A, const float* __restrict__ B,
                 float* __restrict__ Cout, int M, int Kd, int Nd) {
  __shared__ __align__(16) float As[2][128][GKC + 4];  // 36.9 KB
  __shared__ __align__(16) float Wt[2][32][GKC + 4];   //  9.2 KB (Wt[n][k])

  const int tid  = threadIdx.x;
  const int wave = tid >> 5;
  const int lane = tid & 31;
  const int r    = lane & 15;
  const int hh   = lane >> 4;
  const int ncol = Nd >> 5;
  const int bm   = (blockIdx.x / ncol) * 128;
  const int bn   = (blockIdx.x % ncol) * 32;
  const int kk   = tid >> 3;                 // 0..31 (W chunk row)
  const int c4   = (tid & 7) * 4;            // 0..28 (W chunk col group)
  const int nchunks = Kd / GKC;

  v8f acc00 = vzero8(), acc01 = vzero8(), acc10 = vzero8(), acc11 = vzero8();

  // prologue: stage Wt[0] cooperatively, DMA As chunk 0
  {
    float4 wv = *(const float4*)(B + (size_t)kk * Nd + bn + c4);
    Wt[0][c4 + 0][kk] = wv.x;
    Wt[0][c4 + 1][kk] = wv.y;
    Wt[0][c4 + 2][kk] = wv.z;
    Wt[0][c4 + 3][kk] = wv.w;
    if (wave == 0) {
      tdm_load_2d(A + (size_t)bm * Kd, lds_off(&As[0][0][0]),
                  GKC, 128, Kd, M, Kd);
      __builtin_amdgcn_s_wait_tensorcnt(0);
    }
  }
  __syncthreads();

  for (int i = 0; i < nchunks; ++i) {
    const int buf = i & 1;
    const bool havnext = (i + 1 < nchunks);
    float4 wv;
    if (havnext) {                           // W prefetch overlaps compute
      const int kc2 = (i + 1) * GKC;
      wv = *(const float4*)(B + (size_t)(kc2 + kk) * Nd + bn + c4);
      if (wave == 0)
        tdm_load_2d(A + (size_t)bm * Kd + kc2, lds_off(&As[buf ^ 1][0][0]),
                    GKC, 128, Kd, M, Kd);
    }

#pragma unroll
    for (int k4 = 0; k4 < GKC; k4 += 8) {    // 16 WMMAs per staged chunk
      v2f a0  = *(const v2f*)&As[buf][wave * 16 + r][k4 + 2 * hh];
      v2f b00 = *(const v2f*)&Wt[buf][r][k4 + 2 * hh];
      v2f b01 = *(const v2f*)&Wt[buf][16 + r][k4 + 2 * hh];
      acc00 = wmma_f32(a0, b00, acc00);
      acc01 = wmma_f32(a0, b01, acc01);
      v2f a1  = *(const v2f*)&As[buf][wave * 16 + r][k4 + 4 + 2 * hh];
      v2f b10 = *(const v2f*)&Wt[buf][r][k4 + 4 + 2 * hh];
      v2f b11 = *(const v2f*)&Wt[buf][16 + r][k4 + 4 + 2 * hh];
      acc10 = wmma_f32(a1, b10, acc10);
      acc11 = wmma_f32(a1, b11, acc11);
    }

    if (havnext) {                           // store prefetched W to idle buf
      Wt[buf ^ 1][c4 + 0][kk] = wv.x;
      Wt[buf ^ 1][c4 + 1][kk] = wv.y;
      Wt[buf ^ 1][c4 + 2][kk] = wv.z;
      Wt[buf ^ 1][c4 + 3][kk] = wv.w;
    }
    if (wave == 0) __builtin_amdgcn_s_wait_tensorcnt(0);
    __syncthreads();
  }

  v8f c0 = acc00 + acc10;
  v8f c1 = acc01 + acc11;
#pragma unroll
  for (int v = 0; v < 8; ++v) {
    const int m = bm + wave * 16 + 8 * hh + v;
    Cout[(size_t)m * Nd + bn + r]      = c0[v];
    Cout[(size_t)m * Nd + bn + 16 + r] = c1[v];
  }
}

// ----------------------- agg[i] = bias + hw[i]*dinv[i]^2 (self-loop, det.)
__global__ void __launch_bounds__(256)
agg_init_kernel(const float* __restrict__ hw, const float* __restrict__ dinv,
                const float* __restrict__ bias, float* __restrict__ agg) {
  const int idx = blockIdx.x * 256 + threadIdx.x;      // N*512 total
  const int i = idx >> 9;
  const int c = idx & 511;
  const float di = dinv[i];
  agg[idx] = bias[c] + hw[idx] * di * di;
}

// --------------------- scatter: agg[j] += hw[i] * dinv[i]*dinv[j] over edges
__global__ void __launch_bounds__(256)
scatter_kernel(const float* __restrict__ hw, const float* __restrict__ dinv,
               const int* __restrict__ knn_idx, float* __restrict__ agg) {
  const int i = blockIdx.x;                            // source node, 8 edges
  const float di = dinv[i];
  const int c = threadIdx.x;                           // 256 threads, 2 ch each
  const float h0 = hw[(size_t)i * C_OUT + c];
  const float h1 = hw[(size_t)i * C_OUT + c + 256];
  for (int s = 0; s < KNN; ++s) {
    const int j = knn_idx[i * KNN + s];
    const float nrm = di * dinv[j];
    atomicAdd(agg + (size_t)j * C_OUT + c,       h0 * nrm);
    atomicAdd(agg + (size_t)j * C_OUT + c + 256, h1 * nrm);
  }
}

// ------------------------------------------------------------------- ReLU
__global__ void __launch_bounds__(256)
relu_kernel(const float* __restrict__ in, float* __restrict__ out, int total) {
  const int idx = blockIdx.x * 256 + threadIdx.x;
  if (idx < total) out[idx] = fmaxf(in[idx], 0.0f);
}

extern "C" void kernel_launch(void* const* d_in, const int* in_sizes, int n_in,
                              void* d_out, int out_size, void* d_ws, size_t ws_size,
                              hipStream_t stream) {
  (void)in_sizes; (void)n_in; (void)out_size; (void)ws_size;
  const float* x  = (const float*)d_in[0];
  const float* W1 = (const float*)d_in[1];
  const float* b1 = (const float*)d_in[2];
  const float* W2 = (const float*)d_in[3];
  const float* b2 = (const float*)d_in[4];

  float* ws   = (float*)d_ws;
  float* sq   = ws;                                      // N
  float* deg  = ws + N_NODES;                            // N (deg -> dinv)
  int*   knn  = (int*)(ws + 2 * N_NODES);                // N*8
  float* hw   = ws + 2 * N_NODES + N_NODES * KNN;        // N*512
  float* agg  = hw + (size_t)N_NODES * C_OUT;            // N*512

  const int nElem = N_NODES * C_OUT;                     // 3211264, /256 exact
  const int gemmBlocks = (N_NODES / 128) * (C_OUT / 32); // 49*16 = 784

  rowsq_kernel<<<N_NODES, 256, 0, stream>>>(x, sq, deg);
  knn_kernel<<<N_NODES / 16, 256, 0, stream>>>(x, sq, knn, deg);
  dinv_kernel<<<(N_NODES + 255) / 256, 256, 0, stream>>>(deg);

  // layer 1: hw = x @ W1 ; agg = b1 + norm-agg ; h1 = relu (in place in agg)
  gemm_wmma_kernel<<<gemmBlocks, 256, 0, stream>>>(x, W1, hw, N_NODES, C_IN, C_OUT);
  agg_init_kernel<<<nElem / 256, 256, 0, stream>>>(hw, deg, b1, agg);
  scatter_kernel<<<N_NODES, 256, 0, stream>>>(hw, deg, knn, agg);
  relu_kernel<<<nElem / 256, 256, 0, stream>>>(agg, agg, nElem);

  // layer 2: hw = h1 @ W2 ; agg reused after gemm completes (stream ordered)
  gemm_wmma_kernel<<<gemmBlocks, 256, 0, stream>>>(agg, W2, hw, N_NODES, C_OUT, C_OUT);
  agg_init_kernel<<<nElem / 256, 256, 0, stream>>>(hw, deg, b2, agg);
  scatter_kernel<<<N_NODES, 256, 0, stream>>>(hw, deg, knn, agg);
  relu_kernel<<<nElem / 256, 256, 0, stream>>>(agg, (float*)d_out, nElem);
}